// SetEq4to4_75350906241868
// MI455X (gfx1250) — compile-verified
//
#include <hip/hip_runtime.h>
#include <hip/hip_bf16.h>

typedef float v2f __attribute__((ext_vector_type(2)));
typedef float v8f __attribute__((ext_vector_type(8)));

#define NDTOT 16          // n*d = 2*8
#define NSP   1048576     // 32^4
#define P3SZ  32768       // 32^3
#define P2SZ  1024
#define P1SZ  32

// ---------------------------------------------------------------------------
// Pass 1: block = (nd, i). One read of x[nd][i][:,:,:] produces
//   S3_ijk[nd][i][j][k] (sum over l)   via LDS rowsum
//   S3_ijl[nd][i][j][l] (sum over k)   via LDS colsum
//   S3_ikl[nd][i][k][l] (sum over j)   via register float4 accumulation
// ---------------------------------------------------------------------------
__global__ void reduce_pass1(const float* __restrict__ x,
                             float* __restrict__ S3_ijk,
                             float* __restrict__ S3_ijl,
                             float* __restrict__ S3_ikl) {
  __shared__ float rowsum[32];
  __shared__ float colsum[32];
  int b = blockIdx.x;
  int nd = b >> 5, i = b & 31;
  int t = threadIdx.x;
  int k = t >> 3, g = t & 7;                 // thread covers (k, l=4g..4g+3)
  size_t base = (size_t)nd * NSP + (size_t)i * P3SZ;
  float4 acc = make_float4(0.f, 0.f, 0.f, 0.f);
  for (int j = 0; j < 32; ++j) {
    if (t < 32) rowsum[t] = 0.f;
    else if (t < 64) colsum[t - 32] = 0.f;
    __syncthreads();
    float4 v = *(const float4*)(x + base + j * 1024 + k * 32 + g * 4);
    atomicAdd(&rowsum[k], v.x + v.y + v.z + v.w);
    atomicAdd(&colsum[g * 4 + 0], v.x);
    atomicAdd(&colsum[g * 4 + 1], v.y);
    atomicAdd(&colsum[g * 4 + 2], v.z);
    atomicAdd(&colsum[g * 4 + 3], v.w);
    acc.x += v.x; acc.y += v.y; acc.z += v.z; acc.w += v.w;
    __syncthreads();
    if (t < 32)
      S3_ijk[(size_t)nd * P3SZ + (i * 32 + j) * 32 + t] = rowsum[t];
    else if (t < 64)
      S3_ijl[(size_t)nd * P3SZ + (i * 32 + j) * 32 + (t - 32)] = colsum[t - 32];
  }
  float* p = S3_ikl + (size_t)nd * P3SZ + (size_t)i * 1024 + k * 32 + g * 4;
  *(float4*)p = acc;
}

// ---------------------------------------------------------------------------
// Pass 2: block = (nd, j). One read of x -> S3_jkl[nd][j][k][l] (sum over i).
// ---------------------------------------------------------------------------
__global__ void reduce_pass2(const float* __restrict__ x,
                             float* __restrict__ S3_jkl) {
  int b = blockIdx.x;
  int nd = b >> 5, j = b & 31;
  int t = threadIdx.x;
  size_t base = (size_t)nd * NSP + (size_t)j * 1024;
  float4 acc = make_float4(0.f, 0.f, 0.f, 0.f);
  for (int i = 0; i < 32; ++i) {
    float4 v = *(const float4*)(x + base + (size_t)i * P3SZ + t * 4);
    acc.x += v.x; acc.y += v.y; acc.z += v.z; acc.w += v.w;
  }
  *(float4*)(S3_jkl + (size_t)nd * P3SZ + (size_t)j * 1024 + t * 4) = acc;
}

// ---------------------------------------------------------------------------
// Small reductions: S2 (6) and S1 (4) from the S3 tensors. Block = nd.
// ---------------------------------------------------------------------------
__global__ void reduce_small(const float* __restrict__ S3_ijk,
                             const float* __restrict__ S3_ijl,
                             const float* __restrict__ S3_jkl,
                             float* S2_ij, float* S2_ik, float* S2_il,
                             float* S2_jk, float* S2_jl, float* S2_kl,
                             float* S1_i, float* S1_j, float* S1_k, float* S1_l) {
  int nd = blockIdx.x;
  int t = threadIdx.x;
  const float* A = S3_ijk + (size_t)nd * P3SZ;
  const float* B = S3_ijl + (size_t)nd * P3SZ;
  const float* C = S3_jkl + (size_t)nd * P3SZ;
  for (int e = t; e < 1024; e += 256) {
    int p = e >> 5, q = e & 31;
    float s_ij = 0.f, s_ik = 0.f, s_jk = 0.f, s_il = 0.f, s_jl = 0.f, s_kl = 0.f;
    for (int r = 0; r < 32; ++r) {
      s_ij += A[(p * 32 + q) * 32 + r];   // i=p, j=q, sum k
      s_ik += A[(p * 32 + r) * 32 + q];   // i=p, k=q, sum j
      s_jk += A[(r * 32 + p) * 32 + q];   // j=p, k=q, sum i
      s_il += B[(p * 32 + r) * 32 + q];   // i=p, l=q, sum j
      s_jl += B[(r * 32 + p) * 32 + q];   // j=p, l=q, sum i
      s_kl += C[(r * 32 + p) * 32 + q];   // k=p, l=q, sum j
    }
    size_t o = (size_t)nd * P2SZ + e;
    S2_ij[o] = s_ij; S2_ik[o] = s_ik; S2_jk[o] = s_jk;
    S2_il[o] = s_il; S2_jl[o] = s_jl; S2_kl[o] = s_kl;
  }
  if (t < 32) {
    float si = 0.f, sj = 0.f, sk = 0.f, sl = 0.f;
    for (int u = 0; u < 1024; ++u) {
      si += A[t * 1024 + u];
      sj += A[(u >> 5) * 1024 + t * 32 + (u & 31)];
      sk += A[u * 32 + t];
      sl += B[u * 32 + t];
    }
    size_t o = (size_t)nd * P1SZ + t;
    S1_i[o] = si; S1_j[o] = sj; S1_k[o] = sk; S1_l[o] = sl;
  }
}

// ---------------------------------------------------------------------------
// Generic WMMA channel-mix: Y[n][so][P] = sum_t mix(S_t[n][d][P], coefs[:, :, idx_t])
// One 16(M=spatial)x16(N=so, top half padded) tile per wave; K=8 via two
// V_WMMA_F32_16X16X4_F32 chained on the accumulator.
// ---------------------------------------------------------------------------
__global__ void mix_kernel(float* __restrict__ Y,
                           const float* S0, const float* S1p, const float* S2p,
                           const float* S3p, const float* S4p, const float* S5p,
                           int i0, int i1, int i2, int i3, int i4, int i5,
                           int count, int P, const float* __restrict__ coefs) {
  int wave = threadIdx.x >> 5;
  int lane = threadIdx.x & 31;
  int tilesPerN = P >> 4;
  int tilesTotal = 2 * tilesPerN;
  int g = blockIdx.x * 8 + wave;
  if (g >= tilesTotal) return;            // wave-uniform: EXEC stays all-ones
  int n = g / tilesPerN;
  int p0 = (g - n * tilesPerN) << 4;
  int so = lane & 15, half = lane >> 4, m = lane & 15;
  int d0 = 2 * half;
  bool live = so < 8;
  v8f acc = {};
  auto term = [&](const float* s, int col) {
    v2f bv0, bv1;
    bv0.x = live ? coefs[((d0 + 0) * 8 + so) * 69 + col] : 0.f;
    bv0.y = live ? coefs[((d0 + 1) * 8 + so) * 69 + col] : 0.f;
    bv1.x = live ? coefs[((d0 + 4) * 8 + so) * 69 + col] : 0.f;
    bv1.y = live ? coefs[((d0 + 5) * 8 + so) * 69 + col] : 0.f;
    const float* sb = s + (size_t)n * 8 * P + p0 + m;
    v2f a0, a1;
    a0.x = sb[(size_t)(d0 + 0) * P]; a0.y = sb[(size_t)(d0 + 1) * P];
    a1.x = sb[(size_t)(d0 + 4) * P]; a1.y = sb[(size_t)(d0 + 5) * P];
    acc = __builtin_amdgcn_wmma_f32_16x16x4_f32(false, a0, false, bv0,
                                                (short)0, acc, false, false);
    acc = __builtin_amdgcn_wmma_f32_16x16x4_f32(false, a1, false, bv1,
                                                (short)0, acc, false, false);
  };
  term(S0, i0);
  if (count > 1) term(S1p, i1);
  if (count > 2) term(S2p, i2);
  if (count > 3) term(S3p, i3);
  if (count > 4) term(S4p, i4);
  if (count > 5) term(S5p, i5);
  if (live) {
    float* yb = Y + (size_t)(n * 8 + so) * P + p0 + 8 * half;
    *(float4*)yb = make_float4(acc[0], acc[1], acc[2], acc[3]);
    *((float4*)yb + 1) = make_float4(acc[4], acc[5], acc[6], acc[7]);
  }
}

// ---------------------------------------------------------------------------
// Fused output kernel: block = (n, i, j). Stages broadcast terms in LDS,
// then WMMA-mixes x (pair 68) and adds Cconst[so][k] + Ydyn[so][k][l] + bias.
// ---------------------------------------------------------------------------
__global__ void final_kernel(const float* __restrict__ x,
                             const float* __restrict__ coefs,
                             const float* __restrict__ bias,
                             const float* Y3_ijk, const float* Y3_ijl,
                             const float* Y3_ikl, const float* Y3_jkl,
                             const float* Y2_ij, const float* Y2_ik,
                             const float* Y2_il, const float* Y2_jk,
                             const float* Y2_jl, const float* Y2_kl,
                             const float* Y1_i, const float* Y1_j,
                             const float* Y1_k, const float* Y1_l,
                             float* __restrict__ out) {
  __shared__ float Wl[8][32];
  __shared__ float Cc[8][32];
  __shared__ float Ydyn[8192];            // [so][k][l]
  int b = blockIdx.x;
  int n = b >> 10, rem = b & 1023, i = rem >> 5, j = rem & 31;
  int t = threadIdx.x;
  size_t n8 = (size_t)n * 8;
  {
    int so_t = t >> 5, a = t & 31;
    size_t o3 = (n8 + so_t) * P3SZ, o2 = (n8 + so_t) * P2SZ, o1 = (n8 + so_t) * P1SZ;
    Wl[so_t][a] = Y3_ijl[o3 + (i * 32 + j) * 32 + a] + Y2_il[o2 + i * 32 + a]
                + Y2_jl[o2 + j * 32 + a] + Y1_l[o1 + a];
    Cc[so_t][a] = bias[so_t]
                + Y3_ijk[o3 + (i * 32 + j) * 32 + a]
                + Y2_ij[o2 + i * 32 + j] + Y2_ik[o2 + i * 32 + a]
                + Y2_jk[o2 + j * 32 + a]
                + Y1_i[o1 + i] + Y1_j[o1 + j] + Y1_k[o1 + a];
  }
  __syncthreads();
  for (int r = 0; r < 32; ++r) {
    int e = t + r * 256;
    int so = e >> 10, k = (e >> 5) & 31, l = e & 31;
    size_t o3 = (n8 + so) * P3SZ, o2 = (n8 + so) * P2SZ;
    Ydyn[e] = Wl[so][l]
            + Y3_ikl[o3 + (i * 32 + k) * 32 + l]
            + Y3_jkl[o3 + (j * 32 + k) * 32 + l]
            + Y2_kl[o2 + k * 32 + l];
  }
  __syncthreads();
  int wave = t >> 5, lane = t & 31;
  int so = lane & 15, half = lane >> 4, m = lane & 15;
  int d0 = 2 * half;
  bool live = so < 8;
  v2f b0, b1;
  b0.x = live ? coefs[((d0 + 0) * 8 + so) * 69 + 68] : 0.f;
  b0.y = live ? coefs[((d0 + 1) * 8 + so) * 69 + 68] : 0.f;
  b1.x = live ? coefs[((d0 + 4) * 8 + so) * 69 + 68] : 0.f;
  b1.y = live ? coefs[((d0 + 5) * 8 + so) * 69 + 68] : 0.f;
  size_t xn = n8 * NSP;
  size_t plane = (size_t)(i * 32 + j) * 1024;
  for (int q = 0; q < 8; ++q) {
    int T = wave * 8 + q;
    int k = T >> 1, h = T & 1, l0 = h * 16;
    const float* xb = x + xn + plane + k * 32 + l0 + m;
    v2f a0, a1;
    a0.x = xb[(size_t)(d0 + 0) * NSP]; a0.y = xb[(size_t)(d0 + 1) * NSP];
    a1.x = xb[(size_t)(d0 + 4) * NSP]; a1.y = xb[(size_t)(d0 + 5) * NSP];
    v8f acc = {};
    acc = __builtin_amdgcn_wmma_f32_16x16x4_f32(false, a0, false, b0,
                                                (short)0, acc, false, false);
    acc = __builtin_amdgcn_wmma_f32_16x16x4_f32(false, a1, false, b1,
                                                (short)0, acc, false, false);
    if (live) {
      float cbase = Cc[so][k];
      int lb = l0 + 8 * half;
      const float* yd = Ydyn + so * 1024 + k * 32 + lb;
      float r0 = acc[0] + cbase + yd[0];
      float r1 = acc[1] + cbase + yd[1];
      float r2 = acc[2] + cbase + yd[2];
      float r3 = acc[3] + cbase + yd[3];
      float r4 = acc[4] + cbase + yd[4];
      float r5 = acc[5] + cbase + yd[5];
      float r6 = acc[6] + cbase + yd[6];
      float r7 = acc[7] + cbase + yd[7];
      float* ob = out + (n8 + so) * NSP + plane + k * 32 + lb;
      *(float4*)ob = make_float4(r0, r1, r2, r3);
      *((float4*)ob + 1) = make_float4(r4, r5, r6, r7);
    }
  }
}

// ---------------------------------------------------------------------------
extern "C" void kernel_launch(void* const* d_in, const int* in_sizes, int n_in,
                              void* d_out, int out_size, void* d_ws, size_t ws_size,
                              hipStream_t stream) {
  (void)in_sizes; (void)n_in; (void)out_size; (void)ws_size;
  const float* x     = (const float*)d_in[0];
  const float* coefs = (const float*)d_in[1];
  const float* bias  = (const float*)d_in[2];
  float* out = (float*)d_out;

  const size_t P3e = (size_t)NDTOT * P3SZ;   // 524288 floats
  const size_t P2e = (size_t)NDTOT * P2SZ;   // 16384
  const size_t P1e = (size_t)NDTOT * P1SZ;   // 512
  float* w = (float*)d_ws;
  float* S3[4]; for (int t = 0; t < 4; ++t) S3[t] = w + t * P3e;
  float* S2[6]; for (int t = 0; t < 6; ++t) S2[t] = w + 4 * P3e + t * P2e;
  float* S1[4]; for (int t = 0; t < 4; ++t) S1[t] = w + 4 * P3e + 6 * P2e + t * P1e;
  float* Y3[4]; for (int t = 0; t < 4; ++t) Y3[t] = w + 4 * P3e + 6 * P2e + 4 * P1e + t * P3e;
  float* Y2[6]; for (int t = 0; t < 6; ++t) Y2[t] = w + 8 * P3e + 6 * P2e + 4 * P1e + t * P2e;
  float* Y1[4]; for (int t = 0; t < 4; ++t) Y1[t] = w + 8 * P3e + 12 * P2e + 4 * P1e + t * P1e;

  reduce_pass1<<<512, 256, 0, stream>>>(x, S3[0], S3[1], S3[2]);
  reduce_pass2<<<512, 256, 0, stream>>>(x, S3[3]);
  reduce_small<<<16, 256, 0, stream>>>(S3[0], S3[1], S3[3],
                                       S2[0], S2[1], S2[2], S2[3], S2[4], S2[5],
                                       S1[0], S1[1], S1[2], S1[3]);
  // Y3_B = sum over 4 rank-3 A-terms; pair index = 52 + ai*4 + bi
  for (int bi = 0; bi < 4; ++bi)
    mix_kernel<<<512, 256, 0, stream>>>(Y3[bi],
        S3[0], S3[1], S3[2], S3[3], S3[0], S3[0],
        52 + 0 * 4 + bi, 52 + 1 * 4 + bi, 52 + 2 * 4 + bi, 52 + 3 * 4 + bi, 0, 0,
        4, P3SZ, coefs);
  // Y2_B = sum over 6 rank-2 A-terms; pair index = 16 + ai*6 + bi
  for (int bi = 0; bi < 6; ++bi)
    mix_kernel<<<16, 256, 0, stream>>>(Y2[bi],
        S2[0], S2[1], S2[2], S2[3], S2[4], S2[5],
        16 + 0 * 6 + bi, 16 + 1 * 6 + bi, 16 + 2 * 6 + bi,
        16 + 3 * 6 + bi, 16 + 4 * 6 + bi, 16 + 5 * 6 + bi,
        6, P2SZ, coefs);
  // Y1_B = sum over 4 rank-1 A-terms; pair index = ai*4 + bi
  for (int bi = 0; bi < 4; ++bi)
    mix_kernel<<<1, 256, 0, stream>>>(Y1[bi],
        S1[0], S1[1], S1[2], S1[3], S1[0], S1[0],
        0 * 4 + bi, 1 * 4 + bi, 2 * 4 + bi, 3 * 4 + bi, 0, 0,
        4, P1SZ, coefs);

  final_kernel<<<2048, 256, 0, stream>>>(x, coefs, bias,
      Y3[0], Y3[1], Y3[2], Y3[3],
      Y2[0], Y2[1], Y2[2], Y2[3], Y2[4], Y2[5],
      Y1[0], Y1[1], Y1[2], Y1[3],
      out);
}